// EDOCDLoss_10118942949454
// MI455X (gfx1250) — compile-verified
//
#include <hip/hip_runtime.h>
#include <stdint.h>

// ---------------------------------------------------------------------------
// OCD / KL distillation loss, MI455X (gfx1250).
// Dominant cost: streaming 262 MB of `outputs` once -> ~11 us at 23.3 TB/s.
// Stage A: edit-distance DP (tiny) -> distinct optimal-token sets per (b,i).
// Stage B: per-row sum over V via async global->LDS, 4-deep pipeline
//          (s_wait_asynccnt), WMMA partial-sum reduce, <=64-token gather,
//          closed-form KL row loss.
// Stage C: deterministic final reduction to the scalar.
// ---------------------------------------------------------------------------

#define B_ 32
#define P_ 64
#define L_ 64
#define V_ 32000

typedef float v2f __attribute__((ext_vector_type(2)));
typedef float v4f __attribute__((ext_vector_type(4)));
typedef float v8f __attribute__((ext_vector_type(8)));
typedef int   v4i __attribute__((ext_vector_type(4)));

#if __has_builtin(__builtin_amdgcn_global_load_async_to_lds_b128) && \
    __has_builtin(__builtin_amdgcn_s_wait_asynccnt)
#define HAVE_ASYNC 1
#else
#define HAVE_ASYNC 0
#endif
#if __has_builtin(__builtin_amdgcn_wmma_f32_16x16x4_f32)
#define HAVE_WMMA 1
#else
#define HAVE_WMMA 0
#endif

#if defined(__AMDGCN__) && !HAVE_ASYNC
#warning "gfx1250 device pass: async-to-LDS builtins NOT found; using direct NT loads"
#endif
#if defined(__AMDGCN__) && !HAVE_WMMA
#warning "gfx1250 device pass: wmma_f32_16x16x4_f32 builtin NOT found; using tree reduce"
#endif

// Builtin takes int4* pointers: AS(1) source, AS(3) LDS destination.
// Generic->LDS: low 32 bits of a generic shared address are the LDS offset.
typedef __attribute__((address_space(1))) v4i gv4i_t;
typedef __attribute__((address_space(3))) v4i lv4i_t;

__device__ __forceinline__ gv4i_t* to_glob_v4i(const void* p) {
  return (gv4i_t*)(unsigned long long)p;
}
__device__ __forceinline__ lv4i_t* to_lds_v4i(const void* p) {
  return (lv4i_t*)(unsigned int)(unsigned long long)p;
}

// ---------------------------------------------------------------------------
// Stage A: wavefront edit-distance DP. One block per batch, 64 threads (=L).
// Emits, per (b,i): count K and the list of DISTINCT optimal next tokens.
// ---------------------------------------------------------------------------
__global__ __launch_bounds__(64)
void ocd_editdp_kernel(const int* __restrict__ osym,
                       const int* __restrict__ tgt,
                       const unsigned char* __restrict__ mask,
                       int* __restrict__ toks, int* __restrict__ kcnt)
{
  const int b = blockIdx.x;
  const int j = threadIdx.x;

  __shared__ float s_prev[L_];
  __shared__ float s_scan[L_];
  __shared__ float s_red[L_];
  __shared__ int   s_tok[L_];
  __shared__ int   s_opt[L_];
  __shared__ int   s_first[L_];

  const int tj = tgt[b * L_ + j];
  const int mj = mask[b * L_ + j] ? 1 : 0;
  s_tok[j] = tj;
  float prev = (float)j;                       // DP row 0 = arange(L)
  __syncthreads();

  for (int i = 0; i < P_; ++i) {
    float cur;
    if (i == 0) {
      cur = (float)j;
    } else {
      const int   sym = osym[b * P_ + (i - 1)];
      const float pm1 = s_prev[(j == 0) ? 0 : (j - 1)];
      const int   tm1 = s_tok [(j == 0) ? 0 : (j - 1)];
      float tmp;
      if (j == 0) tmp = (float)i;
      else        tmp = fminf(pm1 + ((sym != tm1) ? 1.0f : 0.0f),  // diag (DSUB)
                              prev + 1.0f);                        // insert (DINS)
      // deletion prefix-min: cur[j] = j + cummin_{k<=j}(tmp[k] - k)  (DDEL=1)
      float v = tmp - (float)j;
      s_scan[j] = v;
      __syncthreads();
      for (int d = 1; d < L_; d <<= 1) {
        const float o = (j >= d) ? s_scan[j - d] : v;
        __syncthreads();
        v = fminf(v, o);
        s_scan[j] = v;
        __syncthreads();
      }
      cur = v + (float)j;
    }

    // masked row minimum
    s_red[j] = mj ? cur : __builtin_inff();
    __syncthreads();
    for (int d = L_ / 2; d > 0; d >>= 1) {
      if (j < d) s_red[j] = fminf(s_red[j], s_red[j + d]);
      __syncthreads();
    }
    const float mind = s_red[0];
    const int opt = (mj && (cur == mind)) ? 1 : 0;
    s_opt[j] = opt;
    __syncthreads();

    // distinct-first flag: count each optimal token once
    int first = opt;
    if (opt) {
      for (int q = 0; q < j; ++q)
        if (s_opt[q] && (s_tok[q] == tj)) { first = 0; break; }
    }
    s_first[j] = first;
    __syncthreads();

    if (j == 0) {
      int k = 0;
      const int64_t base = (int64_t)(b * P_ + i) * L_;
      for (int q = 0; q < L_; ++q)
        if (s_first[q]) toks[base + (k++)] = s_tok[q];
      kcnt[b * P_ + i] = k;
    }
    __syncthreads();
    s_prev[j] = cur;
    prev = cur;
    __syncthreads();
  }
}

// ---------------------------------------------------------------------------
// Stage B: one block per (b,i) row. Stream 32000 f32 (125 KB) through LDS
// with a 4-deep async pipeline, reduce with one v_wmma_f32_16x16x4_f32,
// gather the <=64 optimal-token logits, emit the closed-form row loss.
// ---------------------------------------------------------------------------
#define RB_THREADS 256
#define V4_  (V_ / 4)                              // 8000 float4 per row
#define NCH_ ((V4_ + RB_THREADS - 1) / RB_THREADS) // 32 chunks (last partial)
#define ADEPTH 4                                   // async pipeline depth

__global__ __launch_bounds__(RB_THREADS)
void ocd_rowloss_kernel(const float* __restrict__ outputs,
                        const int* __restrict__ toks,
                        const int* __restrict__ kcnt,
                        const unsigned char* __restrict__ mask,
                        float* __restrict__ rloss)
{
  const int row = blockIdx.x;            // b*P + i
  const int b   = row / P_;
  const int i   = row - b * P_;
  const int tid = threadIdx.x;
  const float* __restrict__ rp = outputs + (int64_t)row * V_;

  __shared__ float red[RB_THREADS];
  __shared__ float s_total;

  float acc = 0.0f;
#if HAVE_ASYNC
  {
    __shared__ v4f lbuf[ADEPTH][RB_THREADS];  // 16 KB, 1 float4/lane/buffer
    auto issue = [&](int c) {
      const int g  = tid + c * RB_THREADS;
      const int gc = (g < V4_) ? g : 0;          // clamp tail, weight 0 later
      __builtin_amdgcn_global_load_async_to_lds_b128(
          to_glob_v4i(rp + 4 * gc), to_lds_v4i(&lbuf[c & (ADEPTH - 1)][tid]), 0, 0);
    };
    // prologue: keep 3 chunks in flight ahead of the consumer
    issue(0); issue(1); issue(2);
    for (int k = 0; k < NCH_; ++k) {
      // async loads complete in order, so asynccnt<=N bounds the oldest chunk
      if      (k + 3 < NCH_) { issue(k + 3); __builtin_amdgcn_s_wait_asynccnt(3); }
      else if (k + 2 < NCH_) {               __builtin_amdgcn_s_wait_asynccnt(2); }
      else if (k + 1 < NCH_) {               __builtin_amdgcn_s_wait_asynccnt(1); }
      else                   {               __builtin_amdgcn_s_wait_asynccnt(0); }
      const v4f d = lbuf[k & (ADEPTH - 1)][tid];   // each lane reads its own slot
      const float w = (tid + k * RB_THREADS < V4_) ? 1.0f : 0.0f;
      acc += w * (d.x + d.y + d.z + d.w);
    }
  }
#else
  for (int k = 0; k < NCH_; ++k) {
    const int g = tid + k * RB_THREADS;
    if (g < V4_) {
      const v4f d = __builtin_nontemporal_load(((const v4f*)rp) + g);
      acc += d.x + d.y + d.z + d.w;
    }
  }
#endif

  red[tid] = acc;
  __syncthreads();

#if HAVE_WMMA
  // Wave 0 reduces all 256 partials: 64 coarse partials packed into the f32
  // 16x4 A matrix, B = ones => D rows hold row-sums of A. Summing all D rows
  // (8 VGPRs + one lane-16 swap) yields the total; the mapping of lanes into
  // A is irrelevant because we sum every A entry exactly once. EXEC is all
  // ones within wave 0 (tid<32 covers the full wave), as WMMA requires.
  if (tid < 32) {
    const int base = tid * 8;
    const float q0 = red[base + 0] + red[base + 1] + red[base + 2] + red[base + 3];
    const float q1 = red[base + 4] + red[base + 5] + red[base + 6] + red[base + 7];
    v2f a;  a.x = q0;  a.y = q1;
    v2f bm; bm.x = 1.0f; bm.y = 1.0f;
    v8f c = {};
    v8f d = __builtin_amdgcn_wmma_f32_16x16x4_f32(
        /*neg_a=*/false, a, /*neg_b=*/false, bm,
        /*c_mod=*/(short)0, c, /*reuse_a=*/false, /*reuse_b=*/false);
    float s = d[0] + d[1] + d[2] + d[3] + d[4] + d[5] + d[6] + d[7];
    s += __shfl_xor(s, 16, 32);
    if (tid == 0) s_total = s;
  }
  __syncthreads();
#else
  for (int d = RB_THREADS / 2; d > 0; d >>= 1) {
    if (tid < d) red[tid] += red[tid + d];
    __syncthreads();
  }
  if (tid == 0) s_total = red[0];
  __syncthreads();
#endif

  // gather the distinct optimal-token logits (K <= 64); row is hot in L2
  const int K = kcnt[row];
  if (tid < L_) {
    float gv = 0.0f;
    if (tid < K) gv = rp[toks[(int64_t)row * L_ + tid]];
    red[tid] = gv;
  }
  __syncthreads();

  if (tid == 0) {
    float sumS = 0.0f;
    for (int q = 0; q < L_; ++q) sumS += red[q];
    float lv = 0.0f;
    if (K > 0 && mask[b * L_ + i]) {
      const float e1 = 2.71828182845904523f;
      const float Kf = (float)K;
      const float Z  = Kf * e1 + ((float)V_ - Kf);   // softmax partition
      const float p1 = e1 / Z;
      const float p0 = 1.0f / Z;
      // sum_v p log p = K*p1 - log Z ; sum_v p*out = p0*total + (p1-p0)*sumS
      lv = Kf * p1 - logf(Z) - p0 * s_total - (p1 - p0) * sumS;
    }
    rloss[row] = lv;                                  // deterministic, no atomics
  }
}

// ---------------------------------------------------------------------------
// Stage C: 32 lanes (= B), fixed-order sums -> scalar.
// ---------------------------------------------------------------------------
__global__ __launch_bounds__(32)
void ocd_final_kernel(const float* __restrict__ rloss,
                      const unsigned char* __restrict__ mask,
                      float* __restrict__ out)
{
  const int b = threadIdx.x;
  float ls = 0.0f;
  for (int i = 0; i < P_; ++i) ls += rloss[b * P_ + i];
  float w = 0.0f;
  for (int j = 0; j < L_; ++j) w += mask[b * L_ + j] ? 1.0f : 0.0f;
  float pb = ls / (w + 1e-13f);
  float nz = (w > 0.0f) ? 1.0f : 0.0f;
  for (int d = 16; d > 0; d >>= 1) {
    pb += __shfl_xor(pb, d, 32);
    nz += __shfl_xor(nz, d, 32);
  }
  if (b == 0) out[0] = pb / (nz + 1e-13f);
}

// ---------------------------------------------------------------------------
extern "C" void kernel_launch(void* const* d_in, const int* in_sizes, int n_in,
                              void* d_out, int out_size, void* d_ws, size_t ws_size,
                              hipStream_t stream)
{
  const float*         outputs = (const float*)d_in[0];          // (B,P,V) f32
  const int*           osym    = (const int*)d_in[1];            // (B,P) i32
  const int*           tgt     = (const int*)d_in[2];            // (B,L) i32
  const unsigned char* mask    = (const unsigned char*)d_in[3];  // (B,L) bool (1B)
  float*               out     = (float*)d_out;                  // scalar f32

  // workspace layout: toks[B*P*L] i32 | kcnt[B*P] i32 | rloss[B*P] f32 (~528 KB)
  int*   toks  = (int*)d_ws;
  int*   kcnt  = toks + (size_t)B_ * P_ * L_;
  float* rloss = (float*)(kcnt + B_ * P_);

  ocd_editdp_kernel <<<B_,      L_,         0, stream>>>(osym, tgt, mask, toks, kcnt);
  ocd_rowloss_kernel<<<B_ * P_, RB_THREADS, 0, stream>>>(outputs, toks, kcnt, mask, rloss);
  ocd_final_kernel  <<<1,       32,         0, stream>>>(rloss, mask, out);

  (void)in_sizes; (void)n_in; (void)out_size; (void)ws_size;
}